// LightGCN_74783970557981
// MI455X (gfx1250) — compile-verified
//
#include <hip/hip_runtime.h>

#define NUM_USERS 100000
#define NUM_ITEMS 200000
#define N_NODES   (NUM_USERS + NUM_ITEMS)
#define N_EDGES   6400000
#define DIM       64
#define N_LAYERS  3
#define BATCH     4096

typedef float v2f __attribute__((ext_vector_type(2)));
typedef float v4f __attribute__((ext_vector_type(4)));
typedef float v8f __attribute__((ext_vector_type(8)));

// ---------------------------------------------------------------------------
// cur = concat(user_emb, item_emb); acc = cur; nxt = 0
// ue/ie are read exactly once -> NT loads.  acc is streamed -> NT store.
// cur (gather source) and nxt (atomic destination) stay RT so their lines
// land hot in the 192MB L2.
// ---------------------------------------------------------------------------
__global__ void lgcn_init(const float* __restrict__ ue, const float* __restrict__ ie,
                          float* __restrict__ cur, float* __restrict__ nxt,
                          float* __restrict__ acc) {
    unsigned i = blockIdx.x * blockDim.x + threadIdx.x;
    const unsigned U64 = NUM_USERS * DIM;
    const unsigned T   = N_NODES * DIM;
    if (i < T) {
        float v = (i < U64) ? __builtin_nontemporal_load(ue + i)
                            : __builtin_nontemporal_load(ie + (i - U64));
        cur[i] = v;
        __builtin_nontemporal_store(v, acc + i);
        nxt[i] = 0.0f;
    }
}

// ---------------------------------------------------------------------------
// One SpMM layer: nxt[dst] += val * cur[src].  16 lanes per edge, float4 per
// lane -> gather is a fully-coalesced 256B burst per edge, 4 native f32
// atomics per lane.  Edge triples (76.8MB/layer) are stream-once -> NT loads
// so they don't evict the L2-resident cur/nxt embedding tables (2x76.8MB in
// the 192MB L2).
// ---------------------------------------------------------------------------
__global__ void lgcn_scatter(const float* __restrict__ emb,
                             const float* __restrict__ eval,
                             const int*   __restrict__ esrc,
                             const int*   __restrict__ edst,
                             float*       __restrict__ out) {
    unsigned t = blockIdx.x * blockDim.x + threadIdx.x;
    unsigned e = t >> 4;
    if (e >= N_EDGES) return;
    unsigned c = (t & 15u) * 4u;

    int   s = __builtin_nontemporal_load(esrc + e);
    int   d = __builtin_nontemporal_load(edst + e);
    float w = __builtin_nontemporal_load(eval + e);

    const v4f x = *reinterpret_cast<const v4f*>(emb + (size_t)s * DIM + c);
    float* o = out + (size_t)d * DIM + c;
    unsafeAtomicAdd(o + 0, w * x.x);
    unsafeAtomicAdd(o + 1, w * x.y);
    unsafeAtomicAdd(o + 2, w * x.z);
    unsafeAtomicAdd(o + 3, w * x.w);
}

// ---------------------------------------------------------------------------
// acc += nxt ; zero old cur (it becomes next layer's atomic destination).
// acc is a pure stream -> NT load + NT store.  nxt becomes next layer's
// gather source -> RT read keeps it resident.  Zero-store to oldcur pre-warms
// dirty lines in L2 for next layer's atomics -> RT.
// ---------------------------------------------------------------------------
__global__ void lgcn_post(float* __restrict__ acc, const float* __restrict__ nxt,
                          float* __restrict__ oldcur) {
    unsigned i = blockIdx.x * blockDim.x + threadIdx.x;
    const unsigned T = N_NODES * DIM;
    if (i < T) {
        float a = __builtin_nontemporal_load(acc + i);
        __builtin_nontemporal_store(a + nxt[i], acc + i);
        oldcur[i] = 0.0f;
    }
}

// ---------------------------------------------------------------------------
// Batched dot via V_WMMA_F32_16X16X4_F32.  Each wave handles 16 pairs:
//   D(16x16) = sum_k U(16x4) * I^T(4x16),  gamma = diag(D) / 16
// A layout (f32 16x4): lane m=lane%16 holds K = (lane>>4)*2 + {0,1}
// B layout (f32 4x16): lane n=lane%16 holds K = (lane>>4)*2 + {0,1}
// Diagonal (m,m) sits in c[m%8] of lane m + 16*(m>>3).
// EXEC is all-ones up to and through the WMMAs (no divergence before them).
// ---------------------------------------------------------------------------
__global__ void lgcn_dot_wmma(const float* __restrict__ acc,
                              const int*   __restrict__ users,
                              const int*   __restrict__ items,
                              float*       __restrict__ out) {
    unsigned lane = threadIdx.x & 31u;
    unsigned wave = threadIdx.x >> 5;
    unsigned p0   = (blockIdx.x * (blockDim.x >> 5) + wave) * 16u;

    unsigned m  = lane & 15u;
    unsigned kb = (lane >> 4) * 2u;

    int ur = users[p0 + m];
    int ir = NUM_USERS + items[p0 + m];
    const float* urow = acc + (size_t)ur * DIM;
    const float* irow = acc + (size_t)ir * DIM;

    v8f c = {0.f, 0.f, 0.f, 0.f, 0.f, 0.f, 0.f, 0.f};
#pragma unroll
    for (int kk = 0; kk < DIM / 4; ++kk) {
        v2f a, b;
        a.x = urow[kk * 4 + kb];
        a.y = urow[kk * 4 + kb + 1];
        b.x = irow[kk * 4 + kb];
        b.y = irow[kk * 4 + kb + 1];
        c = __builtin_amdgcn_wmma_f32_16x16x4_f32(
                /*neg_a=*/false, a, /*neg_b=*/false, b,
                /*c_mod=*/(short)0, c, /*reuse_a=*/false, /*reuse_b=*/false);
    }

    // gamma = diag / ((L+1)^2)  (folds the 1/(L+1) scaling of both operands)
    const float scale = 1.0f / ((N_LAYERS + 1) * (N_LAYERS + 1));
#pragma unroll
    for (int v = 0; v < 8; ++v) {
        unsigned mm = (unsigned)v + (lane >> 4) * 8u;
        if ((lane & 15u) == mm) out[p0 + mm] = c[v] * scale;
    }
}

// ---------------------------------------------------------------------------
extern "C" void kernel_launch(void* const* d_in, const int* in_sizes, int n_in,
                              void* d_out, int out_size, void* d_ws, size_t ws_size,
                              hipStream_t stream) {
    const float* ue    = (const float*)d_in[0];
    const float* ie    = (const float*)d_in[1];
    const float* ev    = (const float*)d_in[2];
    const int*   es    = (const int*)d_in[3];
    const int*   ed    = (const int*)d_in[4];
    const int*   users = (const int*)d_in[5];
    const int*   items = (const int*)d_in[6];
    float*       out   = (float*)d_out;

    const size_t T = (size_t)N_NODES * DIM;   // 19.2M floats = 76.8 MB
    float* cur = (float*)d_ws;
    float* nxt = cur + T;
    float* acc = nxt + T;

    const dim3 blk(256);
    const unsigned gElem = (unsigned)((T + 255) / 256);
    const unsigned gScat = (unsigned)(((size_t)N_EDGES * 16 + 255) / 256);

    lgcn_init<<<gElem, blk, 0, stream>>>(ue, ie, cur, nxt, acc);

    for (int l = 0; l < N_LAYERS; ++l) {
        lgcn_scatter<<<gScat, blk, 0, stream>>>(cur, ev, es, ed, nxt);
        lgcn_post<<<gElem, blk, 0, stream>>>(acc, nxt, cur);
        float* tmp = cur; cur = nxt; nxt = tmp;   // nxt (old cur) is zeroed
    }

    // 4096 pairs / (16 per wave * 8 waves per block) = 32 blocks
    lgcn_dot_wmma<<<BATCH / 128, blk, 0, stream>>>(acc, users, items, out);
}